// FESM_36146444763825
// MI455X (gfx1250) — compile-verified
//
#include <hip/hip_runtime.h>
#include <hip/hip_bf16.h>
#include <math.h>

typedef __attribute__((ext_vector_type(2))) float v2f;
typedef __attribute__((ext_vector_type(4))) float v4f;
typedef __attribute__((ext_vector_type(8))) float v8f;

#define NB      32
#define NC      1024
#define SPATIAL 1024          // W*H = 32*32
#define FEPS    1e-5f

// ---------------------------------------------------------------------------
// Kernel 1: conv_w / conv_h = sum_c fm[b,c,pos] * w[c], via V_WMMA_F32_16X16X4_F32
// M = 16 spatial positions, K = channels (4 per step), B-matrix col0 = w_w,
// col1 = w_h, cols 2..15 = 0.  Each wave produces 16 positions of both convs.
// fm is read with default (RT) policy so it stays resident in the 192MB L2
// for the pass-3 re-read.
// ---------------------------------------------------------------------------
__global__ void fesm_conv_wmma(const float* __restrict__ fm,
                               const float* __restrict__ wW,
                               const float* __restrict__ wH,
                               float* __restrict__ convW,
                               float* __restrict__ convH) {
    __shared__ float lds[3 * NC];        // [0]=w_w, [1]=w_h, [2]=zeros
    const int tid = threadIdx.x;
#pragma unroll
    for (int i = tid; i < NC; i += 256) {     // branchless 3-segment fill
        lds[i]          = wW[i];
        lds[NC + i]     = wH[i];
        lds[2 * NC + i] = 0.0f;
    }
    __syncthreads();

    const int lane = tid & 31;
    const int wid  = tid >> 5;
    const int b    = blockIdx.y;
    const int posBase = blockIdx.x * 128 + wid * 16;   // 8 waves * 16 pos = 128/block
    const int m = lane & 15;          // M row (A) == N col (B/D) lane index
    const int g = lane >> 4;          // K-group selector

    const float* wsel = &lds[(m < 2 ? m : 2) * NC];    // col0->w_w, col1->w_h, else 0
    const float* fmB  = fm + (size_t)b * (NC * SPATIAL) + (posBase + m);

    v8f acc = {};
#pragma unroll 4
    for (int k0 = 0; k0 < NC; k0 += 4) {
        const int kc = k0 + 2 * g;
        v2f av, bv;
        av.x = fmB[(size_t)kc * SPATIAL];          // A[m][2g]
        av.y = fmB[(size_t)(kc + 1) * SPATIAL];    // A[m][2g+1]
        bv.x = wsel[kc];                           // B[2g][n]
        bv.y = wsel[kc + 1];                       // B[2g+1][n]
        acc = __builtin_amdgcn_wmma_f32_16x16x4_f32(
                  false, av, false, bv, (short)0, acc, false, false);
    }

    // D layout: VGPR j -> rows j (lanes 0-15) / j+8 (lanes 16-31), col = lane&15
    if (m < 2) {                                    // lanes 0,1,16,17 hold cols 0,1
        float* dst = (m == 0) ? convW : convH;
        const int base = b * SPATIAL + posBase + 8 * g;
#pragma unroll
        for (int j = 0; j < 8; ++j) dst[base + j] = acc[j];
    }
}

// ---------------------------------------------------------------------------
// Kernel 2: BN + ReLU + strip avg-pool + softmax -> 8 factors per batch:
// factors[b][sw][sh] = { 1 + 0.5*aW[sw]*aH[sh],  (f < 0.95*max) ? 1 : 0.5 }
// One block per batch, 1024 threads (one per spatial position).
// ---------------------------------------------------------------------------
__global__ void fesm_attn(const float* __restrict__ convW,
                          const float* __restrict__ convH,
                          const float* __restrict__ bW, const float* __restrict__ gW,
                          const float* __restrict__ btW, const float* __restrict__ mW,
                          const float* __restrict__ vW,
                          const float* __restrict__ bH, const float* __restrict__ gH,
                          const float* __restrict__ btH, const float* __restrict__ mH,
                          const float* __restrict__ vH,
                          float* __restrict__ factors) {
    __shared__ float sm[32 * 4];
    const int tid = threadIdx.x;
    const int b   = blockIdx.x;
    const int pos = tid;
    const int w = pos >> 5, h = pos & 31;
    const int sw = w >> 4,  sh = h >> 4;

    const float scW = gW[0] * rsqrtf(vW[0] + FEPS);
    const float ofW = btW[0] + (bW[0] - mW[0]) * scW;
    const float scH = gH[0] * rsqrtf(vH[0] + FEPS);
    const float ofH = btH[0] + (bH[0] - mH[0]) * scH;

    const float actW = fmaxf(0.0f, convW[b * SPATIAL + pos] * scW + ofW);
    const float actH = fmaxf(0.0f, convH[b * SPATIAL + pos] * scH + ofH);

    float v0 = sw ? 0.0f : actW;
    float v1 = sw ? actW : 0.0f;
    float v2 = sh ? 0.0f : actH;
    float v3 = sh ? actH : 0.0f;
#pragma unroll
    for (int off = 16; off > 0; off >>= 1) {    // wave32 tree reduction
        v0 += __shfl_down(v0, off, 32);
        v1 += __shfl_down(v1, off, 32);
        v2 += __shfl_down(v2, off, 32);
        v3 += __shfl_down(v3, off, 32);
    }
    const int lane = tid & 31, widx = tid >> 5;
    if (lane == 0) {
        sm[widx * 4 + 0] = v0; sm[widx * 4 + 1] = v1;
        sm[widx * 4 + 2] = v2; sm[widx * 4 + 3] = v3;
    }
    __syncthreads();

    if (tid == 0) {
        float s0 = 0, s1 = 0, s2 = 0, s3 = 0;
        for (int i = 0; i < 32; ++i) {
            s0 += sm[i * 4 + 0]; s1 += sm[i * 4 + 1];
            s2 += sm[i * 4 + 2]; s3 += sm[i * 4 + 3];
        }
        const float p0 = s0 * (1.0f / 512.0f), p1 = s1 * (1.0f / 512.0f);
        const float q0 = s2 * (1.0f / 512.0f), q1 = s3 * (1.0f / 512.0f);
        const float mw = fmaxf(p0, p1);
        const float e0 = expf(p0 - mw), e1 = expf(p1 - mw);
        const float aW0 = e0 / (e0 + e1), aW1 = e1 / (e0 + e1);
        const float mh = fmaxf(q0, q1);
        const float f0 = expf(q0 - mh), f1 = expf(q1 - mh);
        const float aH0 = f0 / (f0 + f1), aH1 = f1 / (f0 + f1);
        const float thresh = 0.95f * fmaxf(aW0, aW1) * fmaxf(aH0, aH1);
        const float aW[2] = {aW0, aW1}, aH[2] = {aH0, aH1};
        for (int i = 0; i < 2; ++i)
            for (int j = 0; j < 2; ++j) {
                const float f = aW[i] * aH[j];
                factors[b * 8 + (i * 2 + j) * 2 + 0] = 1.0f + 0.5f * f;
                factors[b * 8 + (i * 2 + j) * 2 + 1] = (f < thresh) ? 1.0f : 0.5f;
            }
    }
}

// ---------------------------------------------------------------------------
// Kernel 3: streaming elementwise pass. 4-float vectors: read fm once (should
// hit L2, still warm from pass 1: 128MB fm < 192MB L2), write boost + suppress
// with NON-TEMPORAL (TH=NT) stores so the 256MB of output does not evict fm
// mid-pass. Strip indices are constant inside a 4-float group (16 % 4 == 0).
// ---------------------------------------------------------------------------
__global__ void fesm_apply(const float* __restrict__ fm,
                           const float* __restrict__ factors,
                           float* __restrict__ out) {
    const long long idx4 = (long long)blockIdx.x * blockDim.x + threadIdx.x;
    const long long i    = idx4 * 4;                  // flat element index
    const int b       = (int)(i >> 20);               // C*W*H = 2^20 per batch
    const int spatial = (int)(i & 1023);
    const int sw = (spatial >> 9) & 1;
    const int sh = (spatial >> 4) & 1;
    const v2f fc = *(const v2f*)&factors[b * 8 + (sw * 2 + sh) * 2];

    const v4f x  = __builtin_nontemporal_load(((const v4f*)fm) + idx4);
    const v4f bo = x * fc.x;                          // boost  = fm*(1+0.5*f)
    const v4f su = x * fc.y;                          // suppress = fm*mask
    __builtin_nontemporal_store(bo, ((v4f*)out) + idx4);
    __builtin_nontemporal_store(su, ((v4f*)out) + 8388608 + idx4);  // 2nd half
}

// ---------------------------------------------------------------------------
extern "C" void kernel_launch(void* const* d_in, const int* in_sizes, int n_in,
                              void* d_out, int out_size, void* d_ws, size_t ws_size,
                              hipStream_t stream) {
    const float* fm  = (const float*)d_in[0];
    const float* wW  = (const float*)d_in[1];
    const float* bW  = (const float*)d_in[2];
    const float* gW  = (const float*)d_in[3];
    const float* btW = (const float*)d_in[4];
    const float* mW  = (const float*)d_in[5];
    const float* vW  = (const float*)d_in[6];
    const float* wH  = (const float*)d_in[7];
    const float* bH  = (const float*)d_in[8];
    const float* gH  = (const float*)d_in[9];
    const float* btH = (const float*)d_in[10];
    const float* mH  = (const float*)d_in[11];
    const float* vH  = (const float*)d_in[12];

    float* ws      = (float*)d_ws;
    float* convW   = ws;                     // 32*1024 floats
    float* convH   = ws + NB * SPATIAL;      // 32*1024 floats
    float* factors = ws + 2 * NB * SPATIAL;  // 32*8 floats

    // Pass 1: 8 blocks (128 positions each) x 32 batches, 256 thr = 8 waves
    fesm_conv_wmma<<<dim3(8, NB), 256, 0, stream>>>(fm, wW, wH, convW, convH);
    // Pass 2: one block per batch
    fesm_attn<<<NB, 1024, 0, stream>>>(convW, convH, bW, gW, btW, mW, vW,
                                       bH, gH, btH, mH, vH, factors);
    // Pass 3: 33,554,432 elems / 4 per thread / 256 per block = 32768 blocks
    fesm_apply<<<32768, 256, 0, stream>>>(fm, factors, (float*)d_out);
}